// LSTM_66718021976236
// MI455X (gfx1250) — compile-verified
//
#include <hip/hip_runtime.h>

typedef __attribute__((ext_vector_type(2))) float v2f;
typedef __attribute__((ext_vector_type(8))) float v8f;

#define SEQ_T 48
#define IN_SZ 5
#define HID   8

// ds_swizzle group-of-32 patterns: offset[4:0]=and, [9:5]=or, [14:10]=xor
#define SWZ_XOR8 0x201F

__device__ __forceinline__ float fsig(float x) {
    // sigmoid via native v_exp_f32 + v_rcp_f32
    return __builtin_amdgcn_rcpf(1.0f + __expf(-x));
}
__device__ __forceinline__ float ftanh(float x) {
    return fmaf(2.0f, fsig(2.0f * x), -1.0f);
}
__device__ __forceinline__ float swz_xor8(float v) {
    return __int_as_float(__builtin_amdgcn_ds_swizzle(__float_as_int(v), SWZ_XOR8));
}

__device__ __forceinline__ v8f wmma4(v2f a, v2f b, v8f c) {
    // D = A(16x4,f32) * B(4x16,f32) + C(16x16,f32)
    return __builtin_amdgcn_wmma_f32_16x16x4_f32(false, a, false, b, (short)0, c,
                                                 false, false);
}

__global__ __launch_bounds__(256) void lstm_wmma_kernel(
    const float* __restrict__ x,    const float* __restrict__ W_ih,
    const float* __restrict__ W_hh, const float* __restrict__ b_ih,
    const float* __restrict__ b_hh, const float* __restrict__ W_fc,
    const float* __restrict__ b_fc, float* __restrict__ out, int Btotal)
{
    __shared__ float hbuf_all[8 * 16 * HID];  // 8 waves * 16 items * H = 4 KB
    const int lane = threadIdx.x & 31;
    const int wave = threadIdx.x >> 5;
    const int b0   = (blockIdx.x * 8 + wave) * 16;   // wave-uniform batch tile base
    if (b0 >= Btotal) return;                        // whole-wave exit: EXEC stays full
    float* hb = &hbuf_all[wave * (16 * HID)];

    const int nlo = lane & 15;              // N (or M) index within 16
    const int k0  = (lane >= 16) ? 2 : 0;   // K offset of upper lane half
    const float m4 = (k0 == 0) ? 1.0f : 0.0f;  // lane mask for K=4 (K=6 padded 0)

    // ---- B matrices, resident in VGPRs for the whole kernel ----
    // layout (f32 4x16): vgpr j, lane l -> K = j + 2*(l>=16), N = l%16
    v2f BxLo0, BxHi0, BxLo1, BxHi1;          // B[k][n] = W_ih[n][k], K=5 padded to 8
    BxLo0[0] = W_ih[nlo * IN_SZ + k0];
    BxLo0[1] = W_ih[nlo * IN_SZ + k0 + 1];
    BxHi0[0] = W_ih[nlo * IN_SZ + 4] * m4;   // K=4 ; K=6 -> 0
    BxHi0[1] = 0.0f;                         // K=5,7 -> 0
    BxLo1[0] = W_ih[(16 + nlo) * IN_SZ + k0];
    BxLo1[1] = W_ih[(16 + nlo) * IN_SZ + k0 + 1];
    BxHi1[0] = W_ih[(16 + nlo) * IN_SZ + 4] * m4;
    BxHi1[1] = 0.0f;

    v2f BhLo0, BhHi0, BhLo1, BhHi1;          // B[k][n] = W_hh[n][k], K=8
    BhLo0[0] = W_hh[nlo * HID + k0];
    BhLo0[1] = W_hh[nlo * HID + k0 + 1];
    BhHi0[0] = W_hh[nlo * HID + 4 + k0];
    BhHi0[1] = W_hh[nlo * HID + 5 + k0];
    BhLo1[0] = W_hh[(16 + nlo) * HID + k0];
    BhLo1[1] = W_hh[(16 + nlo) * HID + k0 + 1];
    BhHi1[0] = W_hh[(16 + nlo) * HID + 4 + k0];
    BhHi1[1] = W_hh[(16 + nlo) * HID + 5 + k0];

    // bias preloaded into the C accumulator (broadcast over M rows)
    const float bs0 = b_ih[nlo]      + b_hh[nlo];
    const float bs1 = b_ih[16 + nlo] + b_hh[16 + nlo];
    v8f Cb0, Cb1;
#pragma unroll
    for (int i = 0; i < 8; ++i) { Cb0[i] = bs0; Cb1[i] = bs1; }

    // ---- init state ----
#pragma unroll
    for (int j = 0; j < 4; ++j) hb[lane * 4 + j] = 0.0f;   // h0 = 0
    float cst[4]  = {0.0f, 0.0f, 0.0f, 0.0f};              // c state, per-lane (M,h) slots
    float acc0[4] = {0.0f, 0.0f, 0.0f, 0.0f};              // fused FC accumulators, class 0
    float acc1[4] = {0.0f, 0.0f, 0.0f, 0.0f};              // class 1

    const float* xr = x + (size_t)(b0 + nlo) * (SEQ_T * IN_SZ);
    const bool sel  = (lane & 8) == 0;
    const int  Mofs = ((lane & 8) ? 4 : 0) + ((lane & 16) ? 8 : 0);
    const int  hfc  = lane & 7;

    // ---- software-pipelined global loads (x row + W_fc column), 1 step ahead ----
    v2f   AxLoN;  float x4N, wfc0N, wfc1N;
    AxLoN[0] = xr[k0];
    AxLoN[1] = xr[k0 + 1];
    x4N      = xr[4];
    wfc0N    = W_fc[hfc];
    wfc1N    = W_fc[SEQ_T * HID + hfc];

    for (int t = 0; t < SEQ_T; ++t) {
        // consume pipelined loads for step t
        v2f AxLo = AxLoN;
        v2f AxHi; AxHi[0] = x4N * m4; AxHi[1] = 0.0f;
        const float wfc0 = wfc0N, wfc1 = wfc1N;

        // issue loads for step t+1 (uniform clamped index: no divergence, in-bounds)
        {
            int tn = (t + 1 < SEQ_T) ? (t + 1) : t;
            const float* xt = xr + tn * IN_SZ;
            AxLoN[0] = xt[k0];
            AxLoN[1] = xt[k0 + 1];
            x4N      = xt[4];
            wfc0N    = W_fc[tn * HID + hfc];
            wfc1N    = W_fc[SEQ_T * HID + tn * HID + hfc];
        }

        // recurrent A from LDS staging (same-wave in-order DS)
        v2f AhLo, AhHi;
        AhLo[0] = hb[nlo * HID + k0];
        AhLo[1] = hb[nlo * HID + k0 + 1];
        AhHi[0] = hb[nlo * HID + 4 + k0];
        AhHi[1] = hb[nlo * HID + 5 + k0];

        // gates[16x32] = bias + x_t @ W_ih^T + h @ W_hh^T   (8 f32 WMMAs)
        v8f g0 = Cb0, g1 = Cb1;
        g0 = wmma4(AxLo, BxLo0, g0);
        g0 = wmma4(AxHi, BxHi0, g0);
        g0 = wmma4(AhLo, BhLo0, g0);
        g0 = wmma4(AhHi, BhHi0, g0);
        g1 = wmma4(AxLo, BxLo1, g1);
        g1 = wmma4(AxHi, BxHi1, g1);
        g1 = wmma4(AhLo, BhLo1, g1);
        g1 = wmma4(AhHi, BhHi1, g1);

        // elementwise LSTM cell: pair i/f (and g/o) sitting xor-8 lanes apart;
        // lane halves take distinct M rows (r vs r+4) so no redundant work
#pragma unroll
        for (int r = 0; r < 4; ++r) {
            float d0r  = g0[r], d0r4 = g0[r + 4];
            float d1r  = g1[r], d1r4 = g1[r + 4];
            float s0   = swz_xor8(d0r);
            float s04  = swz_xor8(d0r4);
            float s1   = swz_xor8(d1r);
            float s14  = swz_xor8(d1r4);
            float iv = sel ? d0r : s04;
            float fv = sel ? s0  : d0r4;
            float gv = sel ? d1r : s14;
            float ov = sel ? s1  : d1r4;

            float is = fsig(iv), fs = fsig(fv);
            float gt = ftanh(gv), os = fsig(ov);
            float cn = fmaf(fs, cst[r], is * gt);
            cst[r] = cn;
            float hn = os * ftanh(cn);

            int M = r + Mofs;                       // batch row within tile
            hb[M * HID + hfc] = hn;                 // stage h for next-step A transpose

            acc0[r] = fmaf(hn, wfc0, acc0[r]);      // fused FC head
            acc1[r] = fmaf(hn, wfc1, acc1[r]);
        }
    }

    // reduce FC partials over h (8-lane groups), then write preds
    const float bf0 = b_fc[0], bf1 = b_fc[1];
#pragma unroll
    for (int r = 0; r < 4; ++r) {
#pragma unroll
        for (int m = 1; m <= 4; m <<= 1) {
            acc0[r] += __shfl_xor(acc0[r], m, 32);
            acc1[r] += __shfl_xor(acc1[r], m, 32);
        }
    }
    if ((lane & 7) == 0) {
#pragma unroll
        for (int r = 0; r < 4; ++r) {
            int item = b0 + r + Mofs;
            out[item * 2 + 0] = acc0[r] + bf0;
            out[item * 2 + 1] = acc1[r] + bf1;
        }
    }
}

extern "C" void kernel_launch(void* const* d_in, const int* in_sizes, int n_in,
                              void* d_out, int out_size, void* d_ws, size_t ws_size,
                              hipStream_t stream) {
    (void)n_in; (void)out_size; (void)d_ws; (void)ws_size;
    const float* x    = (const float*)d_in[0];
    const float* W_ih = (const float*)d_in[1];
    const float* W_hh = (const float*)d_in[2];
    const float* b_ih = (const float*)d_in[3];
    const float* b_hh = (const float*)d_in[4];
    const float* W_fc = (const float*)d_in[5];
    const float* b_fc = (const float*)d_in[6];
    float* out = (float*)d_out;

    int B = in_sizes[0] / (SEQ_T * IN_SZ);      // 131072
    int blocks = (B + 127) / 128;               // 128 items per 256-thread block (8 waves)
    lstm_wmma_kernel<<<blocks, 256, 0, stream>>>(x, W_ih, W_hh, b_ih, b_hh,
                                                 W_fc, b_fc, out, B);
}